// PIoULoss_85358180041189
// MI455X (gfx1250) — compile-verified
//
#include <hip/hip_runtime.h>
#include <cmath>

typedef __attribute__((ext_vector_type(2))) float v2f;
typedef __attribute__((ext_vector_type(8))) float v8f;

#define BLOCK_THREADS 256
#define BLOCK_WAVES 8  // 256 / wave32

// Out-of-range coordinate sentinel: makes at least one rotated-axis distance
// ~1e8, so exp() overflows to +inf and rcp(1+inf) == +0 -> pixel kernel is
// exactly 0 with no masking instructions and EXEC stays all-ones (WMMA-safe).
#define COORD_SENTINEL 3.0e8f

struct BoxP {
  float cy, ca, sa, bw, bh;  // bw = 5*w, bh = 5*h (sigmoid bias terms)
};

// sigmoid(-(10*|u| - 5*s)) = 1 / (1 + exp(10*|u| - 5*s))
__device__ __forceinline__ float soft_axis(float u, float bias) {
  float z = __builtin_fmaf(10.0f, __builtin_fabsf(u), -bias);
  return __builtin_amdgcn_rcpf(1.0f + __expf(z));
}

// tx = dx*cos(a), sx = dx*sin(a) hoisted per-lane (x fixed per lane).
__device__ __forceinline__ float box_eval(const BoxP& b, float tx, float sx,
                                          float yv) {
  float dy = yv - b.cy;
  float t = __builtin_fmaf(dy, b.sa, tx);    //  dx*ca + dy*sa
  float d = __builtin_fmaf(dy, b.ca, -sx);   // -dx*sa + dy*ca
  return soft_axis(t, b.bw) * soft_axis(d, b.bh);
}

__global__ __launch_bounds__(BLOCK_THREADS) void piou_tile_kernel(
    const float* __restrict__ pred, const float* __restrict__ target,
    const float* __restrict__ tmpl, int P, int Pr, float* __restrict__ accum) {
  extern __shared__ float sT[];  // Pr floats; tail padded with sentinel

  const int tid = threadIdx.x;
  for (int i = tid; i < Pr; i += BLOCK_THREADS)
    sT[i] = (i < P) ? tmpl[i] : COORD_SENTINEL;
  __syncthreads();

  const int pair = blockIdx.y;
  const int lane = tid & 31;
  // Wave index is uniform within the wave: pin it to an SGPR so the y-tile
  // loop gets scalar control flow (s_cmp/s_cbranch), no EXEC masking.
  const int wv = __builtin_amdgcn_readfirstlane(tid >> 5);

  // Uniform per-block box parameters (scalar loads).
  const float* pb = pred + pair * 5;
  const float* tb = target + pair * 5;
  BoxP BP, BT;
  const float pcx = pb[0];
  BP.cy = pb[1];
  BP.bw = 5.0f * pb[2];
  BP.bh = 5.0f * pb[3];
  {
    float a = pb[4];
    BP.ca = __cosf(a);
    BP.sa = __sinf(a);
  }
  const float tcx = tb[0];
  BT.cy = tb[1];
  BT.bw = 5.0f * tb[2];
  BT.bh = 5.0f * tb[3];
  {
    float a = tb[4];
    BT.ca = __cosf(a);
    BT.sa = __sinf(a);
  }

  // Per-lane x coordinate: M (for A) and N (for B) are both lane&15.
  const int xm = lane & 15;
  const int hi = lane >> 4;
  const int xi = blockIdx.x * 16 + xm;
  const float xl = sT[min(xi, P - 1)];
  const float xv = (xi < P) ? xl : COORD_SENTINEL;  // one select, hoisted

  // Hoist dx-dependent rotation terms out of the y loop.
  const float dxp = xv - pcx;
  const float txp = dxp * BP.ca;
  const float sxp = dxp * BP.sa;
  const float dxt = xv - tcx;
  const float txt = dxt * BT.ca;
  const float sxt = dxt * BT.sa;

  float areaP = 0.0f, areaT = 0.0f;
  v8f acc = {0.f, 0.f, 0.f, 0.f, 0.f, 0.f, 0.f, 0.f};

  // A (Kp, 16x4 f32): lanes 0-15 hold rows K=0,1; lanes 16-31 hold K=2,3.
  // B (Kt, 4x16 f32): vgpr0 rows K = hi; vgpr1 rows K = hi+2.
  const int nyt = Pr >> 2;                       // y tiles of 4
  const int myTiles = (nyt - wv + (BLOCK_WAVES - 1)) / BLOCK_WAVES;  // scalar
  int yA = (wv << 2) + 2 * hi;
  int yB = (wv << 2) + hi;

#pragma unroll 2
  for (int it = 0; it < myTiles; ++it) {
    const float yA0v = sT[yA];
    const float yA1v = sT[yA + 1];
    const float yB0v = sT[yB];
    const float yB1v = sT[yB + 2];

    v2f A, B;
    A.x = box_eval(BP, txp, sxp, yA0v);
    A.y = box_eval(BP, txp, sxp, yA1v);
    B.x = box_eval(BT, txt, sxt, yB0v);
    B.y = box_eval(BT, txt, sxt, yB1v);

    // Each pixel of this 16x4 tile appears exactly once in A and once in B.
    areaP += A.x + A.y;
    areaT += B.x + B.y;

    // D += A x B : Gram matrix sum_y Kp(x,y)*Kt(x',y); diagonal = inter.
    acc = __builtin_amdgcn_wmma_f32_16x16x4_f32(false, A, false, B, (short)0,
                                                acc, false, false);
    yA += BLOCK_WAVES * 4;
    yB += BLOCK_WAVES * 4;
  }

  // 16x16 f32 C/D layout: lanes 0-15 -> (M=vgpr, N=lane),
  // lanes 16-31 -> (M=vgpr+8, N=lane-16). Diagonal lives in
  // lanes 0-7 (vgpr = lane) and lanes 24-31 (vgpr = lane-24).
  float inter;
  {
    int idx = (lane < 8) ? lane : ((lane >= 24) ? (lane - 24) : 0);
    float dv = acc[idx];
    inter = ((lane < 8) || (lane >= 24)) ? dv : 0.0f;
  }

#pragma unroll
  for (int off = 16; off; off >>= 1) {
    inter += __shfl_xor(inter, off, 32);
    areaP += __shfl_xor(areaP, off, 32);
    areaT += __shfl_xor(areaT, off, 32);
  }
  if (lane == 0) {
    atomicAdd(accum + pair * 4 + 0, inter);
    atomicAdd(accum + pair * 4 + 1, areaP);
    atomicAdd(accum + pair * 4 + 2, areaT);
  }
}

__global__ void piou_finalize_kernel(const float* __restrict__ accum,
                                     float* __restrict__ out, int n) {
  const int lane = threadIdx.x;
  float v = 0.0f;
  for (int i = lane; i < n; i += 32) {
    float inter = accum[i * 4 + 0];
    float ap = accum[i * 4 + 1];
    float at = accum[i * 4 + 2];
    float pi = inter / (ap + at - inter + 1e-6f);
    pi = fminf(fmaxf(pi, 0.1f), 1.0f);
    v += -__logf(pi);
  }
#pragma unroll
  for (int off = 16; off; off >>= 1) v += __shfl_xor(v, off, 32);
  if (lane == 0) out[0] = v / ((float)n + 1e-9f);
}

extern "C" void kernel_launch(void* const* d_in, const int* in_sizes, int n_in,
                              void* d_out, int out_size, void* d_ws,
                              size_t ws_size, hipStream_t stream) {
  (void)n_in;
  (void)out_size;
  (void)ws_size;
  const float* pred = (const float*)d_in[0];
  const float* target = (const float*)d_in[1];
  const float* tmpl = (const float*)d_in[2];
  const int n = in_sizes[0] / 5;        // 32 boxes (cx,cy,w,h,a)
  const int P = in_sizes[2];            // 1224 pixel template length
  const int Pr = (P + 3) & ~3;          // rounded to y-tile (4)

  float* accum = (float*)d_ws;          // n * 4 floats: inter, areaP, areaT
  hipMemsetAsync(accum, 0, (size_t)n * 4 * sizeof(float), stream);

  dim3 grid((P + 15) / 16, n);
  size_t shbytes = (size_t)Pr * sizeof(float);
  piou_tile_kernel<<<grid, BLOCK_THREADS, shbytes, stream>>>(pred, target,
                                                             tmpl, P, Pr,
                                                             accum);
  piou_finalize_kernel<<<1, 32, 0, stream>>>(accum, (float*)d_out, n);
}